// APPNP_14688788152988
// MI455X (gfx1250) — compile-verified
//
#include <hip/hip_runtime.h>
#include <hip/hip_bf16.h>

typedef __attribute__((ext_vector_type(2))) float v2f;
typedef __attribute__((ext_vector_type(8))) float v8f;

#define IN_FEATS 512
#define H1DIM    256
#define H2DIM    64
#define NCLS     40
#define KSTEPS   10

// ---------------------------------------------------------------------------
// WMMA f32 GEMM: C[M,NT] = relu(A[M,KT] @ B[KT,NT] + bias)
// Block: 256 threads (8 waves). Block tile: 128 rows x 64 cols.
// Wave computes a 16x64 strip (4 accumulators of 16x16), K staged 64-wide in LDS.
// B is stored K-pair interleaved in LDS so each lane's B fragment is one
// aligned ds_load_b64 (no VGPR repacking before V_WMMA_F32_16X16X4_F32).
// ---------------------------------------------------------------------------
template <int KT, int NT>
__global__ __launch_bounds__(256)
void gemm_bias_relu_wmma(const float* __restrict__ A,
                         const float* __restrict__ B,
                         const float* __restrict__ bias,
                         float* __restrict__ C,
                         int M) {
  constexpr int KB = 64;             // K block staged in LDS
  constexpr int MT = 128;            // rows per workgroup
  constexpr int NTILE = 64;          // cols per workgroup
  constexpr int LDA = KB + 4;        // A row stride: 32x b64 covers all 64 banks
  constexpr int LDB2 = 2 * NTILE + 32; // B kp-row stride: +32 keeps lane halves on
                                       // disjoint bank halves for b64 reads

  __shared__ float As[MT * LDA];         // 128*68*4  = 34816 B
  __shared__ float Bs[(KB / 2) * LDB2];  // 32*160*4  = 20480 B

  const int tid  = threadIdx.x;      // 0..255
  const int lane = tid & 31;
  const int wave = tid >> 5;         // 0..7
  const int row0 = blockIdx.x * MT;
  const int col0 = blockIdx.y * NTILE;

  v8f acc[4] = {};

  const int mn    = lane & 15;           // M index (A/C) and N index (B/C)
  const int khalf = (lane >> 4) * 2;     // lanes 0-15: K 0,1 ; lanes 16-31: K 2,3
  const int moff  = (lane >> 4) * 8;     // C/D row offset for upper lane half

  for (int kb = 0; kb < KT; kb += KB) {
    // --- stage A tile: MT x KB (8192 f32 -> 8 float4 per thread) ---
#pragma unroll
    for (int i = 0; i < (MT * KB) / (256 * 4); ++i) {
      int linear = (tid + i * 256) * 4;
      int r = linear / KB, c = linear % KB;
      int gr = row0 + r;
      float4 v = make_float4(0.f, 0.f, 0.f, 0.f);
      if (gr < M) v = *(const float4*)(A + (size_t)gr * KT + kb + c);
      *(float4*)(&As[r * LDA + c]) = v;
    }
    // --- stage B tile, K-pair interleaved: Bs[kp*LDB2 + 2n + (k&1)] = B[k][n] ---
#pragma unroll
    for (int i = 0; i < (KB * NTILE) / (256 * 4); ++i) {
      int linear = (tid + i * 256) * 4;
      int r = linear / NTILE, c = linear % NTILE;
      float4 v = *(const float4*)(B + (size_t)(kb + r) * NT + col0 + c);
      float* bp = &Bs[(r >> 1) * LDB2 + (r & 1)];
      bp[2 * (c + 0)] = v.x;
      bp[2 * (c + 1)] = v.y;
      bp[2 * (c + 2)] = v.z;
      bp[2 * (c + 3)] = v.w;
    }
    __syncthreads();

    // --- compute: 16 K-steps of 4, 4 WMMAs each ---
#pragma unroll
    for (int kk = 0; kk < KB; kk += 4) {
      const int kb0 = kk + khalf;        // even; lane-half selects K 0,1 vs 2,3
      const int kp  = kb0 >> 1;
      v2f a = *(const v2f*)(&As[(wave * 16 + mn) * LDA + kb0]);
#pragma unroll
      for (int t = 0; t < 4; ++t) {
        v2f b = *(const v2f*)(&Bs[kp * LDB2 + 2 * (t * 16 + mn)]);
        acc[t] = __builtin_amdgcn_wmma_f32_16x16x4_f32(
            false, a, false, b, (short)0, acc[t], false, false);
      }
    }
    __syncthreads();
  }

  // --- epilogue: bias + relu + store (C/D layout: VGPR r -> row r (+8 upper half)) ---
#pragma unroll
  for (int t = 0; t < 4; ++t) {
    int gc = col0 + t * 16 + mn;
    float bv = bias[gc];
#pragma unroll
    for (int r = 0; r < 8; ++r) {
      int gr = row0 + wave * 16 + moff + r;
      if (gr < M) {
        float v = acc[t][r] + bv;
        C[(size_t)gr * NT + gc] = fmaxf(v, 0.0f);
      }
    }
  }
}

// ---------------------------------------------------------------------------
// Graph preprocessing
// ---------------------------------------------------------------------------
__global__ void zero_f32_kernel(float* __restrict__ p, int n) {
  int i = blockIdx.x * blockDim.x + threadIdx.x;
  if (i < n) p[i] = 0.0f;
}

__global__ void degree_kernel(const int* __restrict__ dst, float* __restrict__ deg, int E) {
  int e = blockIdx.x * blockDim.x + threadIdx.x;
  if (e < E)
    __hip_atomic_fetch_add(&deg[dst[e]], 1.0f, __ATOMIC_RELAXED, __HIP_MEMORY_SCOPE_AGENT);
}

__global__ void norm_kernel(const float* __restrict__ deg, float* __restrict__ nrm, int M) {
  int i = blockIdx.x * blockDim.x + threadIdx.x;
  if (i < M) nrm[i] = rsqrtf(fmaxf(deg[i], 1.0f));
}

// ew = (1-alpha) * norm[src] * norm[dst]   (0.9 folded in once for all K steps)
__global__ void edgeweight_kernel(const int* __restrict__ src, const int* __restrict__ dst,
                                  const float* __restrict__ nrm, float* __restrict__ ew, int E) {
  int e = blockIdx.x * blockDim.x + threadIdx.x;
  if (e < E) ew[e] = 0.9f * nrm[src[e]] * nrm[dst[e]];
}

// next = alpha * h0   (then scatter accumulates 0.9*ew*h[src] on top)
__global__ void init_next_kernel(const float4* __restrict__ h0, float4* __restrict__ next, int n4) {
  int i = blockIdx.x * blockDim.x + threadIdx.x;
  if (i < n4) {
    float4 v = h0[i];
    next[i] = make_float4(0.1f * v.x, 0.1f * v.y, 0.1f * v.z, 0.1f * v.w);
  }
}

// one wave32 per edge; each lane covers 2 of the 64 features (float2 + 2 f32 atomics)
__global__ void scatter_kernel(const float* __restrict__ h,
                               const int* __restrict__ src,
                               const int* __restrict__ dst,
                               const float* __restrict__ ew,
                               float* __restrict__ agg, int E) {
  int wv = (int)((blockIdx.x * (unsigned)blockDim.x + threadIdx.x) >> 5);
  int lane = threadIdx.x & 31;
  if (wv >= E) return;
  int s = src[wv], d = dst[wv];
  float w = ew[wv];
  float2 v = ((const float2*)(h + (size_t)s * H2DIM))[lane];
  float* arow = agg + (size_t)d * H2DIM + lane * 2;
  __hip_atomic_fetch_add(arow,     v.x * w, __ATOMIC_RELAXED, __HIP_MEMORY_SCOPE_AGENT);
  __hip_atomic_fetch_add(arow + 1, v.y * w, __ATOMIC_RELAXED, __HIP_MEMORY_SCOPE_AGENT);
}

// out[M,40] = h[M,64] @ W2[64,40] + b2 ; one node per 64-thread block
__global__ __launch_bounds__(64)
void classifier_kernel(const float* __restrict__ h, const float* __restrict__ W2,
                       const float* __restrict__ b2, float* __restrict__ out, int M) {
  __shared__ float hrow[H2DIM];
  int node = blockIdx.x;
  int t = threadIdx.x;
  hrow[t] = h[(size_t)node * H2DIM + t];
  __syncthreads();
  if (t < NCLS) {
    float acc = b2[t];
#pragma unroll
    for (int k = 0; k < H2DIM; ++k) acc += hrow[k] * W2[k * NCLS + t];
    out[(size_t)node * NCLS + t] = acc;
  }
}

// ---------------------------------------------------------------------------
extern "C" void kernel_launch(void* const* d_in, const int* in_sizes, int n_in,
                              void* d_out, int out_size, void* d_ws, size_t ws_size,
                              hipStream_t stream) {
  const float* features = (const float*)d_in[0];
  const int*   src      = (const int*)d_in[1];
  const int*   dst      = (const int*)d_in[2];
  const float* W0       = (const float*)d_in[3];
  const float* b0       = (const float*)d_in[4];
  const float* W1       = (const float*)d_in[5];
  const float* b1       = (const float*)d_in[6];
  const float* W2       = (const float*)d_in[7];
  const float* b2       = (const float*)d_in[8];
  float* out = (float*)d_out;

  const int M = in_sizes[0] / IN_FEATS;   // 100000 nodes
  const int E = in_sizes[1];              // 3200000 edges

  // workspace layout (H1 region reused for propagation state after GEMM2)
  auto al = [](size_t x) { return (x + 255) & ~(size_t)255; };
  const size_t szH  = (size_t)M * H2DIM * sizeof(float);     // 25.6 MB
  const size_t szC1 = (size_t)M * H1DIM * sizeof(float);     // 102.4 MB
  char* p = (char*)d_ws;
  float* C1   = (float*)p;                                   // H1 activations
  float* buf1 = (float*)p;                                   // aliases dead C1
  float* buf2 = (float*)(p + al(szH));
  float* ew   = (float*)(p + 2 * al(szH));
  float* deg  = (float*)(p + 2 * al(szH) + al((size_t)E * 4));
  float* nrm  = (float*)(p + 2 * al(szH) + al((size_t)E * 4) + al((size_t)M * 4));
  size_t aux_end = 2 * al(szH) + al((size_t)E * 4) + 2 * al((size_t)M * 4);
  size_t h0_off  = al(szC1 > aux_end ? szC1 : aux_end);
  float* H0   = (float*)(p + h0_off);                        // needs ~128 MB total

  const int mblk = (M + 127) / 128;

  // 1) H1 = relu(X @ W0 + b0)
  gemm_bias_relu_wmma<IN_FEATS, H1DIM>
      <<<dim3(mblk, H1DIM / 64), 256, 0, stream>>>(features, W0, b0, C1, M);
  // 2) H0 = relu(H1 @ W1 + b1)
  gemm_bias_relu_wmma<H1DIM, H2DIM>
      <<<dim3(mblk, 1), 256, 0, stream>>>(C1, W1, b1, H0, M);

  // 3) degree -> norm -> edge weights (0.9 folded in)
  zero_f32_kernel<<<(M + 255) / 256, 256, 0, stream>>>(deg, M);
  degree_kernel<<<(E + 255) / 256, 256, 0, stream>>>(dst, deg, E);
  norm_kernel<<<(M + 255) / 256, 256, 0, stream>>>(deg, nrm, M);
  edgeweight_kernel<<<(E + 255) / 256, 256, 0, stream>>>(src, dst, nrm, ew, E);

  // 4) K propagation steps: next = 0.1*h0 ; next += 0.9*ew*h[src] (scatter-add)
  const int n4 = M * H2DIM / 4;
  for (int k = 0; k < KSTEPS; ++k) {
    const float* cur = (k == 0) ? H0 : ((k & 1) ? buf1 : buf2);
    float* next = (k & 1) ? buf2 : buf1;
    init_next_kernel<<<(n4 + 255) / 256, 256, 0, stream>>>((const float4*)H0, (float4*)next, n4);
    scatter_kernel<<<(E + 7) / 8, 256, 0, stream>>>(cur, src, dst, ew, next, E);
  }
  float* Hfin = (KSTEPS & 1) ? buf1 : buf2;   // K=10 -> buf2

  // 5) out = Hfin @ W2 + b2
  classifier_kernel<<<M, 64, 0, stream>>>(Hfin, W2, b2, out, M);
}